// Decoder_20701742366850
// MI455X (gfx1250) — compile-verified
//
#include <hip/hip_runtime.h>
#include <math.h>

typedef float v2f __attribute__((ext_vector_type(2)));
typedef float v8f __attribute__((ext_vector_type(8)));

#define BN_SCALE 0.9999950000374997f   // 1/sqrt(1+1e-5)

// ---------------------------------------------------------------------------
// Generic fused GEMM: out[M,N] = epi( A1[M,K1]*sc1 @ B1[K1,N]
//                                   + A2[row>>rsh2, K2]*sc2 @ B2[K2,N] + bias )
// epi: v = (acc + bias[n]) * postscale ; optional ReLU.
// One wave (32 lanes) owns a 32x16 tile (two 16x16 WMMA M-tiles) so every
// B fragment is reused by two V_WMMA_F32_16X16X4_F32 ops (half the B loads,
// 2-way WMMA ILP).  SC1/SC2 are compile-time so unit scales emit no VALU mul.
// K1,K2 multiples of 4; M multiple of 32; N multiple of 64.
// Fragment layouts (ISA 7.12.2): A lane L reg r = A[L%16][2*(L/16)+r],
// B lane L reg r = B[2*(L/16)+r][L%16], D lane L reg r = D[r+8*(L/16)][L%16].
// ---------------------------------------------------------------------------
template<bool SC1, bool SC2>
__global__ __launch_bounds__(128) void gemm_wmma_f32(
    const float* __restrict__ A1, int ldA1, int K1, float sc1,
    const float* __restrict__ A2, int ldA2, int K2, float sc2, int rsh2,
    const float* __restrict__ B1, const float* __restrict__ B2,
    const float* __restrict__ bias, float postscale, int do_relu,
    float* __restrict__ out, int N)
{
    const int wave = threadIdx.x >> 5;
    const int lane = threadIdx.x & 31;
    const int half = lane >> 4;       // 0 or 1
    const int idx  = lane & 15;       // 0..15
    const int m0   = blockIdx.x * 32;
    const int nt   = blockIdx.y * 4 + wave;
    const int ncol = nt * 16 + idx;

    v8f accA = {};                    // rows m0 .. m0+15
    v8f accB = {};                    // rows m0+16 .. m0+31

    {
        const float* a1rA = A1 + (size_t)(m0 + idx) * ldA1;
        const float* a1rB = A1 + (size_t)(m0 + 16 + idx) * ldA1;
        for (int k = 0; k < K1; k += 4) {
            float2 avA = *(const float2*)(a1rA + k + 2 * half);
            float2 avB = *(const float2*)(a1rB + k + 2 * half);
            v2f b;  b[0] = B1[(size_t)(k + 2 * half)     * N + ncol];
                    b[1] = B1[(size_t)(k + 2 * half + 1) * N + ncol];
            v2f aA; v2f aB;
            if (SC1) { aA[0] = avA.x * sc1; aA[1] = avA.y * sc1;
                       aB[0] = avB.x * sc1; aB[1] = avB.y * sc1; }
            else     { aA[0] = avA.x;       aA[1] = avA.y;
                       aB[0] = avB.x;       aB[1] = avB.y; }
            accA = __builtin_amdgcn_wmma_f32_16x16x4_f32(false, aA, false, b,
                                                         (short)0, accA, false, false);
            accB = __builtin_amdgcn_wmma_f32_16x16x4_f32(false, aB, false, b,
                                                         (short)0, accB, false, false);
        }
    }
    if (K2 > 0) {
        const float* a2rA = A2 + (size_t)((m0 + idx) >> rsh2) * ldA2;
        const float* a2rB = A2 + (size_t)((m0 + 16 + idx) >> rsh2) * ldA2;
        for (int k = 0; k < K2; k += 4) {
            float2 avA = *(const float2*)(a2rA + k + 2 * half);
            float2 avB = *(const float2*)(a2rB + k + 2 * half);
            v2f b;  b[0] = B2[(size_t)(k + 2 * half)     * N + ncol];
                    b[1] = B2[(size_t)(k + 2 * half + 1) * N + ncol];
            v2f aA; v2f aB;
            if (SC2) { aA[0] = avA.x * sc2; aA[1] = avA.y * sc2;
                       aB[0] = avB.x * sc2; aB[1] = avB.y * sc2; }
            else     { aA[0] = avA.x;       aA[1] = avA.y;
                       aB[0] = avB.x;       aB[1] = avB.y; }
            accA = __builtin_amdgcn_wmma_f32_16x16x4_f32(false, aA, false, b,
                                                         (short)0, accA, false, false);
            accB = __builtin_amdgcn_wmma_f32_16x16x4_f32(false, aB, false, b,
                                                         (short)0, accB, false, false);
        }
    }

    const float bv = bias[ncol];
#pragma unroll
    for (int r = 0; r < 8; ++r) {
        const int mA = m0 + r + 8 * half;
        const int mB = mA + 16;
        float vA = (accA[r] + bv) * postscale;
        float vB = (accB[r] + bv) * postscale;
        if (do_relu) { vA = fmaxf(vA, 0.0f); vB = fmaxf(vB, 0.0f); }
        out[(size_t)mA * N + ncol] = vA;
        out[(size_t)mB * N + ncol] = vB;
    }
}

// ---------------------------------------------------------------------------
// Pool stage 2 fused with per-group max:
//   m2[j,n] = relu((m1[j,:]@Wm2[:,n] + bm2[n]) * BN_SCALE),  j in group g (32 rows)
//   ph[g,n] = max_j m2[j,n]
// One wave per (N-tile, group): two 16-row WMMA tiles + shfl_xor(16) combine.
// ---------------------------------------------------------------------------
__global__ __launch_bounds__(32) void gemm_m2_groupmax(
    const float* __restrict__ m1, const float* __restrict__ Wm2,
    const float* __restrict__ bm2, float* __restrict__ ph)
{
    const int lane = threadIdx.x & 31;
    const int half = lane >> 4;
    const int idx  = lane & 15;
    const int nt   = blockIdx.x;          // 0..63
    const int g    = blockIdx.y;          // 0..31
    const int ncol = nt * 16 + idx;

    const float* a0 = m1 + (size_t)(g * 32 + idx) * 512;
    const float* a1 = m1 + (size_t)(g * 32 + 16 + idx) * 512;

    v8f c0 = {};
    v8f c1 = {};
    for (int k = 0; k < 512; k += 4) {
        float2 av0 = *(const float2*)(a0 + k + 2 * half);
        float2 av1 = *(const float2*)(a1 + k + 2 * half);
        v2f b; b[0] = Wm2[(size_t)(k + 2 * half)     * 1024 + ncol];
               b[1] = Wm2[(size_t)(k + 2 * half + 1) * 1024 + ncol];
        v2f aa0; aa0[0] = av0.x; aa0[1] = av0.y;
        v2f aa1; aa1[0] = av1.x; aa1[1] = av1.y;
        c0 = __builtin_amdgcn_wmma_f32_16x16x4_f32(false, aa0, false, b,
                                                   (short)0, c0, false, false);
        c1 = __builtin_amdgcn_wmma_f32_16x16x4_f32(false, aa1, false, b,
                                                   (short)0, c1, false, false);
    }

    const float bv = bm2[ncol];
    float mx = 0.0f;                      // ReLU output is >= 0
#pragma unroll
    for (int r = 0; r < 8; ++r) {
        float v0 = fmaxf((c0[r] + bv) * BN_SCALE, 0.0f);
        float v1 = fmaxf((c1[r] + bv) * BN_SCALE, 0.0f);
        mx = fmaxf(mx, fmaxf(v0, v1));
    }
    mx = fmaxf(mx, __shfl_xor(mx, 16, 32));   // combine M halves (lanes n <-> n+16)
    if (half == 0) ph[(size_t)g * 1024 + ncol] = mx;
}

// ---------------------------------------------------------------------------
// LSTM pointwise update + position head + next decoder input.
// One block per pedestrian row, 128 threads (one per hidden unit).
// gates already contains x@W_ih + h@W_hh + b_ih + b_hh.
// ---------------------------------------------------------------------------
__global__ __launch_bounds__(128) void lstm_pointwise(
    const float* __restrict__ gates, float* __restrict__ c,
    float* __restrict__ h_lstm,
    const float* __restrict__ W_h2p, const float* __restrict__ b_h2p,
    const float* __restrict__ W_sp,  const float* __restrict__ b_sp,
    float* __restrict__ di, float* __restrict__ rels_out)
{
    __shared__ float r0[128];
    __shared__ float r1[128];
    const int row = blockIdx.x;
    const int t   = threadIdx.x;

    const float* gr = gates + (size_t)row * 512;
    const float gi = gr[t];
    const float gf = gr[128 + t];
    const float gg = gr[256 + t];
    const float go = gr[384 + t];
    const float si = 1.0f / (1.0f + __expf(-gi));
    const float sf = 1.0f / (1.0f + __expf(-gf));
    const float so = 1.0f / (1.0f + __expf(-go));

    const size_t hoff = (size_t)row * 128 + t;
    const float cn = sf * c[hoff] + si * tanhf(gg);
    c[hoff] = cn;
    const float hl = so * tanhf(cn);
    h_lstm[hoff] = hl;

    r0[t] = hl * W_h2p[t * 2 + 0];
    r1[t] = hl * W_h2p[t * 2 + 1];
    __syncthreads();
    for (int s = 64; s > 0; s >>= 1) {
        if (t < s) { r0[t] += r0[t + s]; r1[t] += r1[t + s]; }
        __syncthreads();
    }
    const float rel0 = r0[0] + b_h2p[0];
    const float rel1 = r1[0] + b_h2p[1];

    if (t == 0) rels_out[row * 2 + 0] = rel0;
    if (t == 1) rels_out[row * 2 + 1] = rel1;
    if (t < 64) di[(size_t)row * 64 + t] = rel0 * W_sp[t] + rel1 * W_sp[64 + t] + b_sp[t];
}

// ---------------------------------------------------------------------------
// One-time init: copy h0/c0, di0 = last_pos_rel@W_sp+b_sp,
// at_emb = at0 * colsum(Wp_at) + bp_at, b_comb = b_ih + b_hh.
// ---------------------------------------------------------------------------
__global__ __launch_bounds__(128) void init_state(
    const float* __restrict__ lpr, const float* __restrict__ h0,
    const float* __restrict__ c0,  const float* __restrict__ at,
    const float* __restrict__ W_sp, const float* __restrict__ b_sp,
    const float* __restrict__ Wp_at, const float* __restrict__ bp_at,
    const float* __restrict__ b_ih, const float* __restrict__ b_hh,
    float* __restrict__ h, float* __restrict__ c, float* __restrict__ di,
    float* __restrict__ at_emb, float* __restrict__ b_comb)
{
    const int row = blockIdx.x;
    const int t   = threadIdx.x;
    const size_t hoff = (size_t)row * 128 + t;
    h[hoff] = h0[hoff];
    c[hoff] = c0[hoff];
    if (t < 64) {
        const float a = at[row];                 // agent_type[0,row,0]
        float s = 0.0f;
#pragma unroll
        for (int k = 0; k < 6; ++k) s += Wp_at[k * 64 + t];
        at_emb[(size_t)row * 64 + t] = a * s + bp_at[t];
        const float x0 = lpr[row * 2 + 0];
        const float x1 = lpr[row * 2 + 1];
        di[(size_t)row * 64 + t] = x0 * W_sp[t] + x1 * W_sp[64 + t] + b_sp[t];
    }
    if (row < 4) b_comb[row * 128 + t] = b_ih[row * 128 + t] + b_hh[row * 128 + t];
}

// ---------------------------------------------------------------------------
extern "C" void kernel_launch(void* const* d_in, const int* in_sizes, int n_in,
                              void* d_out, int out_size, void* d_ws, size_t ws_size,
                              hipStream_t stream) {
    (void)in_sizes; (void)n_in; (void)out_size; (void)ws_size;

    const float* last_pos_rel = (const float*)d_in[1];
    const float* h0    = (const float*)d_in[2];
    const float* c0    = (const float*)d_in[3];
    const float* at    = (const float*)d_in[6];    // agent_type (8,B,1): use [0,:,0]
    const float* W_sp  = (const float*)d_in[8];
    const float* b_sp  = (const float*)d_in[9];
    const float* W_ih  = (const float*)d_in[10];
    const float* W_hh  = (const float*)d_in[11];
    const float* b_ih  = (const float*)d_in[12];
    const float* b_hh  = (const float*)d_in[13];
    const float* W_h2p = (const float*)d_in[14];
    const float* b_h2p = (const float*)d_in[15];
    const float* Wp_at = (const float*)d_in[20];
    const float* bp_at = (const float*)d_in[21];
    const float* Wm1   = (const float*)d_in[26];
    const float* bm1   = (const float*)d_in[27];
    const float* Wm2   = (const float*)d_in[28];
    const float* bm2   = (const float*)d_in[29];
    const float* Wd1   = (const float*)d_in[30];
    const float* bd1   = (const float*)d_in[31];
    const float* Wd2   = (const float*)d_in[32];
    const float* bd2   = (const float*)d_in[33];

    float* ws     = (float*)d_ws;
    float* h      = ws;                 // 1024*128
    float* c      = ws + 131072;        // 1024*128
    float* hl     = ws + 262144;        // 1024*128  (post-LSTM h)
    float* di     = ws + 393216;        // 1024*64
    float* at_emb = ws + 458752;        // 1024*64
    float* ph     = ws + 524288;        // 32*1024   (per-group pooled)
    float* b_comb = ws + 557056;        // 512
    float* gates  = ws + 557568;        // 1024*512
    float* m1     = ws + 1081856;       // 1024*512
    float* dh1    = ws + 1606144;       // 1024*1024
    float* outp   = (float*)d_out;      // rels: 12*1024*2, then h: 1024*128

    init_state<<<1024, 128, 0, stream>>>(last_pos_rel, h0, c0, at, W_sp, b_sp,
                                         Wp_at, bp_at, b_ih, b_hh,
                                         h, c, di, at_emb, b_comb);

    for (int k = 0; k < 12; ++k) {
        // gates = di@W_ih + h@W_hh + (b_ih+b_hh)              M=1024 K=64+128 N=512
        gemm_wmma_f32<false, false><<<dim3(32, 8), 128, 0, stream>>>(
            di, 64, 64, 1.0f,  h, 128, 128, 1.0f, 0,
            W_ih, W_hh, b_comb, 1.0f, 0, gates, 512);

        lstm_pointwise<<<1024, 128, 0, stream>>>(
            gates, c, hl, W_h2p, b_h2p, W_sp, b_sp, di, outp + k * 2048);

        // m1 = bnrelu(0.05*[hl||at_emb] @ Wm1 + bm1)          M=1024 K=128+64 N=512
        gemm_wmma_f32<true, true><<<dim3(32, 8), 128, 0, stream>>>(
            hl, 128, 128, 0.05f,  at_emb, 64, 64, 0.05f, 0,
            Wm1, Wm1 + 128 * 512, bm1, BN_SCALE, 1, m1, 512);

        // ph[g] = max_j bnrelu(m1 @ Wm2 + bm2)                group max, N=1024
        gemm_m2_groupmax<<<dim3(64, 32), 32, 0, stream>>>(m1, Wm2, bm2, ph);

        // dh1 = bnrelu([hl||ph(group)] @ Wd1 + bd1)           M=1024 K=128+1024 N=1024
        gemm_wmma_f32<false, false><<<dim3(32, 16), 128, 0, stream>>>(
            hl, 128, 128, 1.0f,  ph, 1024, 1024, 1.0f, 5,
            Wd1, Wd1 + 128 * 1024, bd1, BN_SCALE, 1, dh1, 1024);

        // h = bnrelu(dh1 @ Wd2 + bd2)                         M=1024 K=1024 N=128
        gemm_wmma_f32<false, false><<<dim3(32, 2), 128, 0, stream>>>(
            dh1, 1024, 1024, 1.0f,  nullptr, 0, 0, 0.0f, 0,
            Wd2, nullptr, bd2, BN_SCALE, 1, h, 128);
    }

    hipMemcpyAsync(outp + 24576, h, 131072 * sizeof(float),
                   hipMemcpyDeviceToDevice, stream);
}